// FocalLoss_45999099740551
// MI455X (gfx1250) — compile-verified
//
#include <hip/hip_runtime.h>
#include <stdint.h>

#define F_ALPHA 0.25f
#define F_WDELTA 0.2f
#define MAXM 64

typedef __attribute__((ext_vector_type(2))) float v2f;
typedef __attribute__((ext_vector_type(8))) float v8f;

// ---------------------------------------------------------------------------
// Wave32 sum via the matrix engine: D = A(16x4) * ones(4x16).
// A VGPR0 = lane value (K=0 for lanes 0-15, K=2 for lanes 16-31), VGPR1 = 0
//   => D[m,n] = val[m] + val[m+16]  (independent of n)
// D vgpr r in lane L: L<16 -> D[r,L] ; L>=16 -> D[r+8,L-16]
// Sum of 8 D vgprs gives S_lo (lanes 0-15) / S_hi (lanes 16-31);
// one shfl_xor(16) + add yields the full 32-lane sum in every lane.
// Requires EXEC == all ones (call only from converged code).
// ---------------------------------------------------------------------------
__device__ __forceinline__ float wave_sum_wmma(float v) {
    v2f a; a[0] = v;    a[1] = 0.0f;
    v2f o; o[0] = 1.0f; o[1] = 1.0f;
    v8f c = {};
    v8f d = __builtin_amdgcn_wmma_f32_16x16x4_f32(
        /*neg_a=*/false, a, /*neg_b=*/false, o,
        /*c_mod=*/(short)0, c, /*reuse_a=*/false, /*reuse_b=*/false);
    float s = ((d[0] + d[1]) + (d[2] + d[3])) + ((d[4] + d[5]) + (d[6] + d[7]));
    s += __shfl_xor(s, 16, 32);
    return s;
}

// ---------------------------------------------------------------------------
// Main kernel: grid = (blocks_per_image, B). Annotations are async-copied to
// LDS (CDNA5 ASYNCcnt path), preprocessed (invalid boxes -> degenerate
// sentinel so IoU==0, removing the per-box validity select), and hoisted to
// registers for the unrolled IoU loop. Regressions load only for positives.
// ---------------------------------------------------------------------------
template <int MT>
__global__ __launch_bounds__(256) void focal_main(
    const float* __restrict__ cls, const float* __restrict__ reg,
    const float* __restrict__ anc, const float* __restrict__ ann,
    float* __restrict__ ws_cls, float* __restrict__ ws_reg,
    unsigned* __restrict__ ws_cnt, int A, int Mdyn)
{
    const int M = (MT > 0) ? MT : (Mdyn > MAXM ? MAXM : Mdyn);
    const int b = blockIdx.y;
    const int tid = threadIdx.x;

    __shared__ float s_ann[MAXM * 5];                 // async staging
    __shared__ __align__(16) float4 s_box[MAXM];      // sentinel-encoded boxes
    __shared__ float s_area[MAXM];
    __shared__ float s_lab[MAXM];

    // --- async copy of this image's annotations into LDS (CDNA5 path) ---
    const float* annb = ann + (size_t)b * M * 5;
    if (tid < M * 5) {
        uint32_t lds_addr = (uint32_t)(uintptr_t)(&s_ann[tid]);  // low 32 bits of flat = LDS offset
        const float* gsrc = annb + tid;
        asm volatile("global_load_async_to_lds_b32 %0, %1, off"
                     :: "v"(lds_addr), "v"(gsrc) : "memory");
    }
    asm volatile("s_wait_asynccnt 0x0" ::: "memory");
    __syncthreads();

    if (tid < M) {
        float x1 = s_ann[tid * 5 + 0], y1 = s_ann[tid * 5 + 1];
        float x2 = s_ann[tid * 5 + 2], y2 = s_ann[tid * 5 + 3];
        float lb = s_ann[tid * 5 + 4];
        bool valid = (lb != -1.0f);
        const float BIG = 3.0e38f;                    // degenerate: IoU == 0 exactly
        s_box[tid]  = valid ? make_float4(x1, y1, x2, y2)
                            : make_float4(BIG, BIG, BIG, BIG);
        s_area[tid] = valid ? (x2 - x1) * (y2 - y1) : 0.0f;
        s_lab[tid]  = lb;
    }
    __syncthreads();

    // --- hoist boxes into registers for the compile-time-M fast path ---
    constexpr int NR = (MT > 0) ? MT : 1;
    float rx1[NR], ry1[NR], rx2[NR], ry2[NR], rar[NR];
    if constexpr (MT > 0) {
        #pragma unroll
        for (int j = 0; j < MT; ++j) {
            float4 bb = s_box[j];
            rx1[j] = bb.x; ry1[j] = bb.y; rx2[j] = bb.z; ry2[j] = bb.w;
            rar[j] = s_area[j];
        }
    }

    const float* clsb = cls + (size_t)b * A * 2;
    const float* regb = reg + (size_t)b * A * 4;

    float acc_c = 0.0f, acc_r = 0.0f;
    float cntf = 0.0f;

    const int stride = gridDim.x * blockDim.x;
    for (int i = blockIdx.x * blockDim.x + tid; i < A; i += stride) {
        float4 an = reinterpret_cast<const float4*>(anc)[i];     // global_load_b128
        float aw = an.z - an.x, ah = an.w - an.y;
        float area_a = aw * ah;

        // IoU vs M boxes: max-only loop (argmax deferred to the rare pos path)
        float iou[NR > 1 ? NR : MAXM];
        float best;
        if constexpr (MT > 0) {
            #pragma unroll
            for (int j = 0; j < MT; ++j) {
                float iw = fmaxf(fminf(an.z, rx2[j]) - fmaxf(an.x, rx1[j]), 0.0f);
                float ih = fmaxf(fminf(an.w, ry2[j]) - fmaxf(an.y, ry1[j]), 0.0f);
                float inter = iw * ih;
                float ua = fmaxf(area_a + rar[j] - inter, 1e-8f);
                iou[j] = inter * __builtin_amdgcn_rcpf(ua);      // v_rcp_f32
            }
            best = iou[0];
            #pragma unroll
            for (int j = 1; j < MT; ++j) best = fmaxf(best, iou[j]);
        } else {
            best = -2.0f;
            for (int j = 0; j < M; ++j) {
                float4 bb = s_box[j];
                float iw = fmaxf(fminf(an.z, bb.z) - fmaxf(an.x, bb.x), 0.0f);
                float ih = fmaxf(fminf(an.w, bb.w) - fmaxf(an.y, bb.y), 0.0f);
                float inter = iw * ih;
                float ua = fmaxf(area_a + s_area[j] - inter, 1e-8f);
                iou[j] = inter * __builtin_amdgcn_rcpf(ua);
                best = fmaxf(best, iou[j]);
            }
        }

        bool pos  = best >= 0.5f;
        bool care = pos || (best < 0.4f);

        // argmax recovery only where it matters (fmax returns an input => exact)
        int bi = 0, lab = -1;
        if (pos) {
            if constexpr (MT > 0) {
                bi = MT - 1;
                #pragma unroll
                for (int j = MT - 2; j >= 0; --j) bi = (iou[j] == best) ? j : bi;
            } else {
                bi = M - 1;
                for (int j = M - 2; j >= 0; --j) bi = (iou[j] == best) ? j : bi;
            }
            lab = (int)s_lab[bi];
        }

        // ------------------- focal classification loss -------------------
        float2 pc = reinterpret_cast<const float2*>(clsb)[i];    // global_load_b64
        if (care) {
            float p0 = fminf(fmaxf(pc.x, 1e-4f), 1.0f - 1e-4f);
            float p1 = fminf(fmaxf(pc.y, 1e-4f), 1.0f - 1e-4f);
            float c0 = (lab == 0)
                ? F_ALPHA * (1.0f - p0) * (1.0f - p0) * (-__logf(p0))
                : (1.0f - F_ALPHA) * p0 * p0 * (-__logf(1.0f - p0));
            float c1 = (lab == 1)
                ? F_ALPHA * (1.0f - p1) * (1.0f - p1) * (-__logf(p1))
                : (1.0f - F_ALPHA) * p1 * p1 * (-__logf(1.0f - p1));
            acc_c += c0 + c1;
        }

        // ------------------- smooth-L1 regression loss -------------------
        if (pos) {                                   // sparse: most reg lines never touched
            cntf += 1.0f;
            float4 rg = reinterpret_cast<const float4*>(regb)[i];
            float4 bb = s_box[bi];                   // dynamic index -> LDS
            float gw = bb.z - bb.x, gh = bb.w - bb.y;
            float gcx = bb.x + 0.5f * gw, gcy = bb.y + 0.5f * gh;
            gw = fmaxf(gw, 1.0f); gh = fmaxf(gh, 1.0f);
            float acx = an.x + 0.5f * aw, acy = an.y + 0.5f * ah;
            float t0 = ((gcx - acx) / aw) / 0.1f;
            float t1 = ((gcy - acy) / ah) / 0.1f;
            float t2 = __logf(gw / aw) / 0.2f;
            float t3 = __logf(gh / ah) / 0.2f;
            float d0 = fabsf(t0 - rg.x), d1 = fabsf(t1 - rg.y);
            float d2 = fabsf(t2 - rg.z), d3 = fabsf(t3 - rg.w);
            const float beta = 1.0f / 9.0f;
            float r0 = (d0 <= beta) ? 4.5f * d0 * d0 : d0 - 0.5f / 9.0f;
            float r1 = (d1 <= beta) ? 4.5f * d1 * d1 : d1 - 0.5f / 9.0f;
            float r2 = (d2 <= beta) ? 4.5f * d2 * d2 : d2 - 0.5f / 9.0f;
            float r3 = (d3 <= beta) ? 4.5f * d3 * d3 : d3 - 0.5f / 9.0f;
            float w = (lab == 0) ? (1.0f + F_WDELTA)
                    : (lab == 1) ? (1.0f - F_WDELTA) : 1.0f;
            acc_r += w * (r0 + r1 + r2 + r3);
        }
    }

    // --- matrix-engine wave reduction (converged here: EXEC all ones) ---
    float wc = wave_sum_wmma(acc_c);
    float wr = wave_sum_wmma(acc_r);
    float wn = wave_sum_wmma(cntf);          // integer-valued, exact in f32
    if ((tid & 31) == 0) {
        atomicAdd(&ws_cls[b], wc);
        atomicAdd(&ws_reg[b], wr);
        atomicAdd(&ws_cnt[b], (unsigned)(wn + 0.5f));
    }
}

// ---------------------------------------------------------------------------
// Finalize: per-image normalization + mean over B, write 2 floats.
// ---------------------------------------------------------------------------
__global__ void focal_final(const float* __restrict__ ws_cls,
                            const float* __restrict__ ws_reg,
                            const unsigned* __restrict__ ws_cnt,
                            float* __restrict__ out, int B)
{
    int t = threadIdx.x;
    float c = 0.0f, r = 0.0f;
    for (int i = t; i < B; i += 32) {
        float np = (float)ws_cnt[i];
        c += ws_cls[i] / fmaxf(np, 1.0f);
        r += (np > 0.0f) ? ws_reg[i] / fmaxf(np * 4.0f, 1.0f) : 0.0f;
    }
    #pragma unroll
    for (int off = 16; off > 0; off >>= 1) {
        c += __shfl_down(c, off, 32);
        r += __shfl_down(r, off, 32);
    }
    if (t == 0) {
        out[0] = c / (float)B;
        out[1] = r / (float)B;
    }
}

extern "C" void kernel_launch(void* const* d_in, const int* in_sizes, int n_in,
                              void* d_out, int out_size, void* d_ws, size_t ws_size,
                              hipStream_t stream) {
    const float* cls = (const float*)d_in[0];   // (B, A, 2)
    const float* reg = (const float*)d_in[1];   // (B, A, 4)
    const float* anc = (const float*)d_in[2];   // (1, A, 4)
    const float* ann = (const float*)d_in[3];   // (B, M, 5)

    const int A = in_sizes[2] / 4;
    const int B = in_sizes[0] / (A * 2);
    const int M = in_sizes[3] / (B * 5);

    float*    ws_cls = (float*)d_ws;
    float*    ws_reg = ws_cls + B;
    unsigned* ws_cnt = (unsigned*)(ws_reg + B);

    hipMemsetAsync(d_ws, 0, (size_t)B * 3 * sizeof(float), stream);

    dim3 grid(128, B);
    if (M == 8) {
        focal_main<8><<<grid, 256, 0, stream>>>(cls, reg, anc, ann,
                                                ws_cls, ws_reg, ws_cnt, A, M);
    } else {
        focal_main<0><<<grid, 256, 0, stream>>>(cls, reg, anc, ann,
                                                ws_cls, ws_reg, ws_cnt, A, M);
    }
    focal_final<<<1, 32, 0, stream>>>(ws_cls, ws_reg, ws_cnt, (float*)d_out, B);
}